// NewNet_1675037246184
// MI455X (gfx1250) — compile-verified
//
#include <hip/hip_runtime.h>
#include <hip/hip_bf16.h>
#include <math.h>

typedef __attribute__((ext_vector_type(2))) float v2f;
typedef __attribute__((ext_vector_type(8))) float v8f;

#define S_DIM 1000
#define C_DIM 100
#define L_DIM 50
#define B_DIM 20
#define K_DIM 64
#define CPAD  112   // 7*16, zero-padded Qt columns

// ---------------------------------------------------------------------------
// Kernel 1: compute Qt[b][c] = Q[c][b] = sum_k Pm[k][c] * (Dm @ P_C)[k][b]
// Pm = softmax over C of (P*P_A+P_B); Dm = softmax over K of (D*D_A+D_B).
// Tiny (~1 MFLOP): one workgroup, plain VALU. Output padded [20][112] in ws.
// ---------------------------------------------------------------------------
__global__ __launch_bounds__(256)
void qmat_kernel(const float* __restrict__ P, const float* __restrict__ D,
                 const float* __restrict__ PA, const float* __restrict__ PB,
                 const float* __restrict__ PC, const float* __restrict__ DA,
                 const float* __restrict__ DB, float* __restrict__ Qt)
{
    __shared__ float Pm[K_DIM][C_DIM];   // 25.6 KB
    __shared__ float Dm[K_DIM][B_DIM];   // 5.1 KB
    __shared__ float Tm[K_DIM][B_DIM];   // 5.1 KB
    __shared__ float PCs[B_DIM][B_DIM];  // 1.6 KB

    const int tid = threadIdx.x;

    // Pm: row-wise softmax over C (one thread per k-row)
    if (tid < K_DIM) {
        const int k = tid;
        float mx = -INFINITY;
        for (int c = 0; c < C_DIM; ++c) {
            float v = P[k * C_DIM + c] * PA[k * C_DIM + c] + PB[k * C_DIM + c];
            Pm[k][c] = v;
            mx = fmaxf(mx, v);
        }
        float sum = 0.f;
        for (int c = 0; c < C_DIM; ++c) { float e = __expf(Pm[k][c] - mx); Pm[k][c] = e; sum += e; }
        float inv = __builtin_amdgcn_rcpf(sum);
        for (int c = 0; c < C_DIM; ++c) Pm[k][c] *= inv;
    }
    // Dm raw values + P_C staging
    for (int i = tid; i < K_DIM * B_DIM; i += 256)
        Dm[i / B_DIM][i % B_DIM] = D[i] * DA[i] + DB[i];
    for (int i = tid; i < B_DIM * B_DIM; i += 256)
        PCs[i / B_DIM][i % B_DIM] = PC[i];
    __syncthreads();

    // Dm: column-wise softmax over K (one thread per b-column)
    if (tid < B_DIM) {
        const int b = tid;
        float mx = -INFINITY;
        for (int k = 0; k < K_DIM; ++k) mx = fmaxf(mx, Dm[k][b]);
        float sum = 0.f;
        for (int k = 0; k < K_DIM; ++k) { float e = __expf(Dm[k][b] - mx); Dm[k][b] = e; sum += e; }
        float inv = __builtin_amdgcn_rcpf(sum);
        for (int k = 0; k < K_DIM; ++k) Dm[k][b] *= inv;
    }
    __syncthreads();

    // Tm = Dm @ P_C   [64x20]
    for (int i = tid; i < K_DIM * B_DIM; i += 256) {
        const int k = i / B_DIM, b = i % B_DIM;
        float t = 0.f;
        for (int j = 0; j < B_DIM; ++j) t += Dm[k][j] * PCs[j][b];
        Tm[k][b] = t;
    }
    __syncthreads();

    // Qt[b][c] = sum_k Pm[k][c] * Tm[k][b], zero-padded to c < 112
    for (int i = tid; i < B_DIM * CPAD; i += 256) {
        const int b = i / CPAD, c = i % CPAD;
        float q = 0.f;
        if (c < C_DIM)
            for (int k = 0; k < K_DIM; ++k) q += Pm[k][c] * Tm[k][b];
        Qt[i] = q;
    }
}

// ---------------------------------------------------------------------------
// Kernel 2 (the bandwidth kernel): 16 samples per block, 4 waves.
// Stage X via GLOBAL_LOAD_ASYNC_TO_LDS_B128 (memory->LDS DMA, ASYNCcnt).
// Phase 1: 320 (sample,b) pairs, single-pass online softmax over L fused with
//          the X-weighted dot -> A_s[16][20] in LDS. csa gathered exactly once.
// Phase 2: per-wave WMMA f32 16x16x4 over K=B_DIM=20 (5 steps) against Qt,
//          fast-sigmoid affine epilogue, store [16 x 100] tile.
// ---------------------------------------------------------------------------
__global__ __launch_bounds__(128)
void fused_kernel(const float* __restrict__ X, const int* __restrict__ stu,
                  const int* __restrict__ cour, const float* __restrict__ csa,
                  const float* __restrict__ Qt, const float* __restrict__ guess_,
                  const float* __restrict__ slide_, float* __restrict__ out)
{
    __shared__ float A_s[16][B_DIM];                  // epilogue GEMM A operand
    __shared__ __align__(16) float X_s[16][L_DIM];    // staged X rows (3.2 KB)

    const int tid = threadIdx.x;
    const long n0 = (long)blockIdx.x * 16;

    // Stage X[n0..n0+15][0..49]: 800 floats = 200 x b128, async global->LDS.
    // Generic LDS pointer low 32 bits == wave-relative LDS byte offset.
    {
        const unsigned lds_base = (unsigned)(uintptr_t)(&X_s[0][0]);
        const float* gsrc = X + n0 * L_DIM;
        for (int i = tid; i < (16 * L_DIM) / 4; i += 128) {
            const unsigned lds_addr = lds_base + (unsigned)i * 16u;
            const unsigned long long ga =
                (unsigned long long)(uintptr_t)(gsrc + (long)i * 4);
            asm volatile("global_load_async_to_lds_b128 %0, %1, off"
                         :: "v"(lds_addr), "v"(ga) : "memory");
        }
        asm volatile("s_wait_asynccnt 0x0" ::: "memory");
    }
    __syncthreads();

    // Phase 1: online softmax + weighted accumulation, one global pass
    for (int p = tid; p < 16 * B_DIM; p += 128) {
        const int s = p / B_DIM;
        const int b = p % B_DIM;
        const long n = n0 + s;
        const long base = ((long)stu[n] * C_DIM + (long)cour[n]) * (L_DIM * B_DIM);
        const float* __restrict__ row = csa + base + b;
        __builtin_prefetch(row, 0, 1);   // global_prefetch hint for the gather
        float m = -INFINITY, sum = 0.f, acc = 0.f;
        #pragma unroll 5
        for (int l = 0; l < L_DIM; ++l) {
            const float v = row[l * B_DIM];
            const float mn = fmaxf(m, v);
            const float r  = __expf(m - mn);
            const float e  = __expf(v - mn);
            sum = sum * r + e;
            acc = acc * r + X_s[s][l] * e;
            m = mn;
        }
        A_s[s][b] = acc * __builtin_amdgcn_rcpf(sum);
    }
    __syncthreads();

    // Phase 2: Y[16 x 100] = A_s[16 x 20] * Qt[20 x 112] via fp32 WMMA
    const int wave  = tid >> 5;
    const int lane  = tid & 31;
    const int mrow  = lane & 15;         // A row / B col / D col within tile
    const int khalf = (lane >> 4) * 2;   // lanes 0-15: K k0,k0+1; 16-31: k0+2,k0+3

    for (int t = wave; t < 7; t += 4) {  // uniform per wave -> EXEC all ones
        const int c0 = t * 16;
        v8f cacc = {};
        #pragma unroll
        for (int k0 = 0; k0 < B_DIM; k0 += 4) {
            v2f a, bf;
            a.x  = A_s[mrow][k0 + khalf];
            a.y  = A_s[mrow][k0 + khalf + 1];
            bf.x = Qt[(k0 + khalf) * CPAD + c0 + mrow];
            bf.y = Qt[(k0 + khalf + 1) * CPAD + c0 + mrow];
            cacc = __builtin_amdgcn_wmma_f32_16x16x4_f32(
                /*neg_a=*/false, a, /*neg_b=*/false, bf,
                /*c_mod=*/(short)0, cacc, /*reuse_a=*/false, /*reuse_b=*/false);
        }
        // Epilogue: (1-slide)*y + guess*(1-y) = y*(1-slide-guess) + guess
        const int c = c0 + mrow;                 // D col = lane & 15
        const int rbase = (lane >> 4) * 8;       // VGPR i -> row rbase + i
        if (c < C_DIM) {
            const float g  = __builtin_amdgcn_rcpf(1.f + __expf(-guess_[c]));
            const float sl = __builtin_amdgcn_rcpf(1.f + __expf(-slide_[c]));
            const float coef = 1.f - sl - g;
            #pragma unroll
            for (int i = 0; i < 8; ++i) {
                const long n = n0 + rbase + i;
                out[n * C_DIM + c] = cacc[i] * coef + g;
            }
        }
    }
}

extern "C" void kernel_launch(void* const* d_in, const int* in_sizes, int n_in,
                              void* d_out, int out_size, void* d_ws, size_t ws_size,
                              hipStream_t stream) {
    const float* X     = (const float*)d_in[0];
    const int*   stu   = (const int*)  d_in[1];
    const int*   cour  = (const int*)  d_in[2];
    const float* csa   = (const float*)d_in[3];
    const float* P     = (const float*)d_in[4];
    const float* D     = (const float*)d_in[5];
    const float* PA    = (const float*)d_in[6];
    const float* PB    = (const float*)d_in[7];
    const float* PC    = (const float*)d_in[8];
    const float* DA    = (const float*)d_in[9];
    const float* DB    = (const float*)d_in[10];
    const float* guess = (const float*)d_in[11];
    const float* slide = (const float*)d_in[12];

    float* Qt = (float*)d_ws;            // [20][112] = 8960 bytes
    const int N = in_sizes[1];           // 16384
    const int nblocks = (N + 15) / 16;   // 1024

    qmat_kernel<<<1, 256, 0, stream>>>(P, D, PA, PB, PC, DA, DB, Qt);
    fused_kernel<<<nblocks, 128, 0, stream>>>(X, stu, cour, csa, Qt, guess, slide,
                                              (float*)d_out);
}